// OVDDetector_15264313770059
// MI455X (gfx1250) — compile-verified
//
#include <hip/hip_runtime.h>
#include <hip/hip_bf16.h>

#define B_  4
#define N_  4096
#define C_  64
#define TARGET_    560.0f
#define MIN_BOX_   5.0f
#define IOU_THR_   0.2f
#define CONF_THR_  0.001f
#define BOXC_THR_  0.01f
#define MAX_WH_    4096.0f

typedef unsigned int u32;
typedef __attribute__((ext_vector_type(4))) u32 u32x4;
typedef __attribute__((ext_vector_type(8))) int i32x8;

// ---------------------------------------------------------------------------
// Kernel 1: clip boxes, per-box class max/argmax, validity, sort key.
// ---------------------------------------------------------------------------
__global__ __launch_bounds__(256) void filter_kernel(
    const float* __restrict__ boxes,      // [B*N,4]
    const float* __restrict__ bscore,     // [B*N]
    const float* __restrict__ clsp,       // [B*N,C]
    float* __restrict__ cbox,             // [B*N,4] clipped
    float* __restrict__ conf,             // [B*N]
    int*   __restrict__ cls,              // [B*N]
    float* __restrict__ key)              // [B*N] sort key (desc)
{
    int gid = blockIdx.x * blockDim.x + threadIdx.x;
    if (gid >= B_ * N_) return;

    float4 bx = *reinterpret_cast<const float4*>(boxes + (size_t)gid * 4);
    float x1 = fminf(fmaxf(bx.x, 0.0f), TARGET_);
    float y1 = fminf(fmaxf(bx.y, 0.0f), TARGET_);
    float x2 = fminf(fmaxf(bx.z, 0.0f), TARGET_);
    float y2 = fminf(fmaxf(bx.w, 0.0f), TARGET_);
    float w = x2 - x1, h = y2 - y1;
    float bs = bscore[gid];

    const float* cp = clsp + (size_t)gid * C_;
    __builtin_prefetch(cp, 0, 1);   // global_prefetch_b8 on gfx1250

    // first-max argmax over C=64 (strict > preserves jnp.argmax tie rule)
    float m = -1e30f; int am = 0;
#pragma unroll
    for (int c = 0; c < C_; c += 4) {
        float4 v = *reinterpret_cast<const float4*>(cp + c);
        if (v.x > m) { m = v.x; am = c + 0; }
        if (v.y > m) { m = v.y; am = c + 1; }
        if (v.z > m) { m = v.z; am = c + 2; }
        if (v.w > m) { m = v.w; am = c + 3; }
    }
    float cf = bs * m;   // bs >= 0, so max(bs*p) == bs*max(p)
    bool valid = (bs > BOXC_THR_) && (w > MIN_BOX_) && (h > MIN_BOX_) && (cf > CONF_THR_);

    float4 cb = make_float4(x1, y1, x2, y2);
    *reinterpret_cast<float4*>(cbox + (size_t)gid * 4) = cb;
    conf[gid] = cf;
    cls[gid]  = am;
    key[gid]  = valid ? cf : -1.0f;
}

// ---------------------------------------------------------------------------
// Kernel 2: per-batch in-LDS bitonic sort (descending by key), then gather
// sorted class-offset boxes, det rows and valid flags to workspace.
// ---------------------------------------------------------------------------
__global__ __launch_bounds__(1024) void sort_gather_kernel(
    const float* __restrict__ cbox,   // [B*N,4]
    const float* __restrict__ conf,   // [B*N]
    const int*   __restrict__ cls,    // [B*N]
    const float* __restrict__ key,    // [B*N]
    float* __restrict__ sob,          // [B*N,4] sorted offset boxes
    float* __restrict__ sdet,         // [B*N,6] sorted det rows
    int*   __restrict__ sval)         // [B*N]   sorted valid flags
{
    __shared__ float sk[N_];
    __shared__ int   si[N_];
    const int b = blockIdx.x, t = threadIdx.x;
    const float* kb = key + (size_t)b * N_;

    for (int k = t; k < N_; k += 1024) { sk[k] = kb[k]; si[k] = k; }
    __syncthreads();

    // bitonic sort, descending
    for (int len = 2; len <= N_; len <<= 1) {
        for (int j = len >> 1; j > 0; j >>= 1) {
            for (int i = t; i < N_; i += 1024) {
                int ixj = i ^ j;
                if (ixj > i) {
                    bool up = ((i & len) == 0);   // up => descending run
                    float a = sk[i], c = sk[ixj];
                    bool sw = up ? (a < c) : (a > c);
                    if (sw) {
                        sk[i] = c; sk[ixj] = a;
                        int tmp = si[i]; si[i] = si[ixj]; si[ixj] = tmp;
                    }
                }
            }
            __syncthreads();
        }
    }

    for (int r = t; r < N_; r += 1024) {
        int o = si[r];
        size_t go = (size_t)b * N_ + o;
        size_t gr = (size_t)b * N_ + r;
        float4 cb = *reinterpret_cast<const float4*>(cbox + go * 4);
        float  cf = conf[go];
        int    cl = cls[go];
        float  off = (float)cl * MAX_WH_;
        float4 ob = make_float4(cb.x + off, cb.y + off, cb.z + off, cb.w + off);
        *reinterpret_cast<float4*>(sob + gr * 4) = ob;
        float* d = sdet + gr * 6;
        d[0] = cb.x; d[1] = cb.y; d[2] = cb.z; d[3] = cb.w;
        d[4] = cf;   d[5] = (float)cl;
        sval[gr] = (sk[r] > 0.0f) ? 1 : 0;
    }
}

// ---------------------------------------------------------------------------
// Kernel 3: per-batch greedy NMS.
//   * 64 KB offset-box table: ONE tensor_load_to_lds (TDM, TENSORcnt) issued
//     by wave 0 — DMA engine moves the tile, waves issue zero vmem for it.
//   * 16 KB valid-flag table: async global->LDS B128 (ASYNCcnt).
//   * suppression mask in LDS; barriers elided on non-keep iterations
//     (keep_i is workgroup-uniform).
// ---------------------------------------------------------------------------
__global__ __launch_bounds__(1024) void nms_kernel(
    const float* __restrict__ sob,    // [B*N,4]
    const float* __restrict__ sdet,   // [B*N,6]
    const int*   __restrict__ sval,   // [B*N]
    float* __restrict__ out)          // [B*N,6]
{
    __shared__ float         lob[N_ * 4];   // 64 KB (TDM destination)
    __shared__ int           lval[N_];      // 16 KB (async destination)
    __shared__ unsigned char lsup[N_];      // 4 KB
    const int b = blockIdx.x, t = threadIdx.x;
    const float* gob  = sob  + (size_t)b * N_ * 4;
    const int*   gval = sval + (size_t)b * N_;

    // ---- TDM: 1-D tile, 16384 x 4-byte elements, global -> LDS -------------
    if (t == 0) {
        unsigned long long ga = (unsigned long long)(uintptr_t)gob;
        unsigned ldsa = (unsigned)(uintptr_t)(&lob[0]);
        u32x4 g0;
        g0.x = 1u;                                        // count=1, user mode
        g0.y = ldsa;                                      // lds_addr
        g0.z = (u32)(ga & 0xFFFFFFFFu);                   // global_addr[31:0]
        g0.w = (u32)((ga >> 32) & 0x01FFFFFFu) | (2u << 30); // addr[56:32] | type=2
        i32x8 g1;
        g1[0] = (int)(2u << 16);          // wg_mask=0 (no cluster), data_size=4B
        g1[1] = (int)((16384u & 0xFFFFu) << 16);          // tensor_dim0 lo16
        g1[2] = (int)(((16384u >> 16) & 0xFFFFu) | (1u << 16)); // dim0 hi | dim1 lo=1
        g1[3] = (int)((16384u & 0xFFFFu) << 16);          // dim1 hi=0 | tile_dim0
        g1[4] = 0;                                        // tile_dim1=0, tile_dim2=0
        g1[5] = 16384;                                    // tensor_dim0_stride lo32
        g1[6] = 0;                                        // stride0 hi | stride1 lo
        g1[7] = 0;                                        // stride1 hi
        asm volatile("tensor_load_to_lds %0, %1" :: "s"(g0), "s"(g1) : "memory");
    }

    // ---- async copy valid flags: one B128 (4 ints) per thread ---------------
    {
        int k4 = t * 4;                                   // 1024 threads * 4 = 4096
        unsigned loff = (unsigned)(uintptr_t)(&lval[k4]);
        unsigned long long ga = (unsigned long long)(uintptr_t)(gval + k4);
        asm volatile("global_load_async_to_lds_b128 %0, %1, off"
                     :: "v"(loff), "v"(ga) : "memory");
    }
    for (int k = t; k < N_; k += 1024) lsup[k] = 0;

    asm volatile("s_wait_asynccnt 0" ::: "memory");       // drain ASYNCcnt
    if (t == 0) __builtin_amdgcn_s_wait_tensorcnt(0);     // drain TENSORcnt (wave 0)
    __syncthreads();

    bool wrote = false;
    for (int i = 0; i < N_; ++i) {
        if (wrote) __syncthreads();                       // uniform condition
        bool keep_i = lval[i] && !lsup[i];
        wrote = false;
        if (keep_i) {
            float4 bi = *reinterpret_cast<const float4*>(&lob[i * 4]);
            float areai = (bi.z - bi.x) * (bi.w - bi.y);
            for (int j = i + 1 + t; j < N_; j += 1024) {
                if (!lsup[j]) {
                    float4 bj = *reinterpret_cast<const float4*>(&lob[j * 4]);
                    float xx1 = fmaxf(bi.x, bj.x), yy1 = fmaxf(bi.y, bj.y);
                    float xx2 = fminf(bi.z, bj.z), yy2 = fminf(bi.w, bj.w);
                    float w = fmaxf(xx2 - xx1, 0.0f), h = fmaxf(yy2 - yy1, 0.0f);
                    float inter = w * h;
                    float areaj = (bj.z - bj.x) * (bj.w - bj.y);
                    float iou = inter / (areai + areaj - inter + 1e-9f);
                    if (iou > IOU_THR_) lsup[j] = 1;
                }
            }
            wrote = true;
        }
    }
    __syncthreads();

    const float* det = sdet + (size_t)b * N_ * 6;
    float* ob = out + (size_t)b * N_ * 6;
    for (int r = t; r < N_; r += 1024) {
        float kmask = (lval[r] && !lsup[r]) ? 1.0f : 0.0f;
#pragma unroll
        for (int c = 0; c < 6; ++c) ob[r * 6 + c] = det[r * 6 + c] * kmask;
    }
}

// ---------------------------------------------------------------------------
extern "C" void kernel_launch(void* const* d_in, const int* in_sizes, int n_in,
                              void* d_out, int out_size, void* d_ws, size_t ws_size,
                              hipStream_t stream) {
    const float* boxes  = (const float*)d_in[0];
    const float* bscore = (const float*)d_in[1];
    const float* clsp   = (const float*)d_in[2];
    float* out = (float*)d_out;

    char* ws = (char*)d_ws;
    const size_t BN = (size_t)B_ * N_;
    float* cbox = (float*)ws;  ws += BN * 4 * sizeof(float);
    float* conf = (float*)ws;  ws += BN * sizeof(float);
    float* key  = (float*)ws;  ws += BN * sizeof(float);
    int*   cls  = (int*)ws;    ws += BN * sizeof(int);
    float* sob  = (float*)ws;  ws += BN * 4 * sizeof(float);
    float* sdet = (float*)ws;  ws += BN * 6 * sizeof(float);
    int*   sval = (int*)ws;

    filter_kernel<<<(int)((BN + 255) / 256), 256, 0, stream>>>(
        boxes, bscore, clsp, cbox, conf, cls, key);
    sort_gather_kernel<<<B_, 1024, 0, stream>>>(
        cbox, conf, cls, key, sob, sdet, sval);
    nms_kernel<<<B_, 1024, 0, stream>>>(sob, sdet, sval, out);
}